// CustomConv1d_6150393168147
// MI455X (gfx1250) — compile-verified
//
#include <hip/hip_runtime.h>

// ---------------------------------------------------------------------------
// Problem constants (mirrors the reference)
// ---------------------------------------------------------------------------
#define C_CH    256
#define KTOP    16.0f
#define ALPHA_L 0.01f
#define N_ITER  50
#define BATCH   8
#define LENGTH  16384
#define N_ROWS  (BATCH * C_CH)          // 2048 independent (b,c) rows
#define TPB     256                     // threads per block (8 wave32)
#define V4_PER_THREAD (LENGTH / (TPB * 4))   // 16 float4 chunks per thread

// ---------------------------------------------------------------------------
// Kernel 1: Dykstra soft top-k on alpha (n=256, 50 iters) in a single wave32,
// then scale[c] = y[0] * V[0, c].  Tiny; latency-irrelevant.
// ---------------------------------------------------------------------------
__global__ void dykstra_scale_kernel(const float* __restrict__ alpha,
                                     const float* __restrict__ V,
                                     float* __restrict__ scale) {
    const int tid = threadIdx.x;            // 0..31, one wave
    float y[8], p[8], q[8];
#pragma unroll
    for (int j = 0; j < 8; ++j) {
        y[j] = alpha[tid * 8 + j] / ALPHA_L;   // s / l
        p[j] = 0.0f;
        q[j] = 0.0f;
    }

    for (int it = 0; it < N_ITER; ++it) {
        float u[8];
        float local = 0.0f;
#pragma unroll
        for (int j = 0; j < 8; ++j) {
            u[j] = y[j] + p[j];
            local += u[j];
        }
        // wave32 reduction for sum(u) over all 256 elements
#pragma unroll
        for (int m = 16; m >= 1; m >>= 1)
            local += __shfl_xor(local, m, 32);
        const float corr = (local - KTOP) * (1.0f / 256.0f);  // (sum(u)-k)/n
#pragma unroll
        for (int j = 0; j < 8; ++j) {
            const float z = u[j] - corr;     // hyperplane projection
            p[j] = corr;                     // p = u - z  (uniform)
            const float v = z + q[j];
            const float yn = fminf(fmaxf(v, 0.0f), 1.0f);  // box projection
            q[j] = v - yn;
            y[j] = yn;
        }
    }

    // only alpha_topk[0] feeds the weights: d[c] = y0 * V[0, c]
    const float y0 = __shfl(y[0], 0, 32);
#pragma unroll
    for (int j = 0; j < 8; ++j) {
        const int e = tid * 8 + j;
        scale[e] = y0 * V[e];                // V[0, e] = first row of [256,256]
    }
}

// ---------------------------------------------------------------------------
// Kernel 2: per-row 3-tap box filter, HBM-bound streaming.
// One block per (b,c) row of 16384 floats.  The whole row is staged in LDS
// via CDNA5 async global->LDS copies (ASYNCcnt path), so each x element is
// pulled from HBM exactly once; zero padding means no cross-block halo.
// ---------------------------------------------------------------------------
__global__ void __launch_bounds__(TPB)
conv_rows_kernel(const float* __restrict__ x,
                 const float* __restrict__ scale,
                 const float* __restrict__ bias,
                 float* __restrict__ out) {
    __shared__ float row[LENGTH];           // 64 KB of the 320 KB WGP LDS

    const int rowid = blockIdx.x;           // 0..2047 == b*256 + c
    const int c     = rowid & (C_CH - 1);
    const float s   = scale[c];
    const float bb  = bias[c];

    const float* __restrict__ xr = x   + (size_t)rowid * LENGTH;
    float*       __restrict__ orw = out + (size_t)rowid * LENGTH;

    // LDS byte offset of the staging buffer (low 32 bits of the shared-
    // aperture flat address == workgroup-relative LDS offset).
    const unsigned lds_base = (unsigned)(size_t)(&row[0]);

    // Issue 16 async B128 copies per thread: 256 thr * 16 * 16 B = 64 KB.
#pragma unroll
    for (int j = 0; j < V4_PER_THREAD; ++j) {
        const unsigned off = (unsigned)(j * TPB + threadIdx.x) * 16u;
        const unsigned lds = lds_base + off;
        asm volatile("global_load_async_to_lds_b128 %0, %1, %2"
                     :: "v"(lds), "v"(off), "s"(xr)
                     : "memory");
    }
    // Wait for this wave's async copies, then sync all 8 waves of the block.
    asm volatile("s_wait_asynccnt 0" ::: "memory");
    __syncthreads();

    // out[t] = s*(x[t-1] + x[t] + x[t+1]) + bias ; zero pad at row ends.
#pragma unroll
    for (int j = 0; j < V4_PER_THREAD; ++j) {
        const int t = (j * TPB + threadIdx.x) * 4;
        const float4 xc = *(const float4*)&row[t];
        const float xm = (t == 0)            ? 0.0f : row[t - 1];
        const float xp = (t + 4 == LENGTH)   ? 0.0f : row[t + 4];
        float4 o;
        o.x = fmaf(s, xm   + xc.x + xc.y, bb);
        o.y = fmaf(s, xc.x + xc.y + xc.z, bb);
        o.z = fmaf(s, xc.y + xc.z + xc.w, bb);
        o.w = fmaf(s, xc.z + xc.w + xp,   bb);
        *(float4*)&orw[t] = o;
    }
}

// ---------------------------------------------------------------------------
// Launch
// ---------------------------------------------------------------------------
extern "C" void kernel_launch(void* const* d_in, const int* in_sizes, int n_in,
                              void* d_out, int out_size, void* d_ws, size_t ws_size,
                              hipStream_t stream) {
    const float* x     = (const float*)d_in[0];   // [8, 256, 16384]
    const float* V     = (const float*)d_in[1];   // [256, 256]
    const float* alpha = (const float*)d_in[2];   // [256]
    const float* bias  = (const float*)d_in[3];   // [256]
    float* out   = (float*)d_out;                 // [8, 256, 16384]
    float* scale = (float*)d_ws;                  // 256 floats of scratch

    dykstra_scale_kernel<<<1, 32, 0, stream>>>(alpha, V, scale);
    conv_rows_kernel<<<N_ROWS, TPB, 0, stream>>>(x, scale, bias, out);
}